// asymmetricLoss_13529146982643
// MI455X (gfx1250) — compile-verified
//
#include <hip/hip_runtime.h>

// CDNA5 / gfx1250, wave32. Streaming weighted-L1 reduction:
//   out = mean( |w*x - w*y| ),  w = (x<y) ? 0.7 : 0.3   (ALPHA = 0.3)
// Memory bound: 268 MB @ 23.3 TB/s ~ 11.5 us floor. NT b128 loads, grid-stride,
// wave-level reduction done with V_WMMA_F32_16X16X4_F32 (A = partials, B = ones).

typedef __attribute__((ext_vector_type(2))) float v2f;
typedef __attribute__((ext_vector_type(4))) float vf4;
typedef __attribute__((ext_vector_type(8))) float v8f;

#define ALPHA_W 0.3f

__device__ __forceinline__ float wterm(float x, float y) {
    float w = (x < y) ? (1.0f - ALPHA_W) : ALPHA_W;   // |0.3 - m|
    return fabsf(w * x - w * y);                      // matches reference rounding
}

__global__ __launch_bounds__(256) void asym_l1_stage1(
    const float* __restrict__ x, const float* __restrict__ y,
    float* __restrict__ partials, long long n)
{
    const long long n4     = n >> 2;
    const long long tid    = (long long)blockIdx.x * blockDim.x + threadIdx.x;
    const long long stride = (long long)gridDim.x * blockDim.x;
    const vf4* __restrict__ x4 = (const vf4*)x;
    const vf4* __restrict__ y4 = (const vf4*)y;

    // Two accumulators -> the two A-matrix VGPRs of the 16x4 f32 WMMA.
    float a0 = 0.0f, a1 = 0.0f;
    for (long long i = tid; i < n4; i += stride) {
        vf4 xv = __builtin_nontemporal_load(&x4[i]);   // global_load_b128 th:NT
        vf4 yv = __builtin_nontemporal_load(&y4[i]);
        a0 += wterm(xv.x, yv.x) + wterm(xv.z, yv.z);
        a1 += wterm(xv.y, yv.y) + wterm(xv.w, yv.w);
    }
    if (tid == 0) {                                    // scalar tail (n % 4)
        for (long long i = (n4 << 2); i < n; ++i)
            a0 += wterm(x[i], y[i]);
    }

    // Wave-level 64-way reduction with one WMMA: D = A(16x4 partials) * B(ones) + 0.
    // B == all ones => layout-agnostic; every D[m][n] = rowsum_m(A).
    // D layout (f32 16x16 C/D): VGPR j, lane 0 -> D[j][0]; lane 16 -> D[j+8][0].
    // So sum_j D_j on lanes 0 and 16 = sum over all 64 A elements.
    v2f a; a.x = a0;   a.y = a1;
    v2f b; b.x = 1.0f; b.y = 1.0f;
    v8f c = {};
    c = __builtin_amdgcn_wmma_f32_16x16x4_f32(false, a, false, b,
                                              (short)0, c, false, false);
    float t = c[0] + c[1] + c[2] + c[3] + c[4] + c[5] + c[6] + c[7];

    __shared__ float lds[16];                          // 8 waves * 2 halves
    const int lane = threadIdx.x & 31;
    const int wave = threadIdx.x >> 5;
    if (lane == 0)  lds[wave * 2 + 0] = t;             // rows 0..7  column 0
    if (lane == 16) lds[wave * 2 + 1] = t;             // rows 8..15 column 0
    __syncthreads();
    if (threadIdx.x == 0) {
        float s = 0.0f;
        #pragma unroll
        for (int i = 0; i < 16; ++i) s += lds[i];
        partials[blockIdx.x] = s;
    }
}

__global__ __launch_bounds__(256) void asym_l1_stage2(
    const float* __restrict__ partials, int nPartials,
    float* __restrict__ out, float invN)
{
    __shared__ float lds[256];
    float s = 0.0f;
    for (int i = threadIdx.x; i < nPartials; i += 256)
        s += partials[i];
    lds[threadIdx.x] = s;
    __syncthreads();
    #pragma unroll
    for (int off = 128; off > 0; off >>= 1) {
        if ((int)threadIdx.x < off) lds[threadIdx.x] += lds[threadIdx.x + off];
        __syncthreads();
    }
    if (threadIdx.x == 0) out[0] = lds[0] * invN;
}

extern "C" void kernel_launch(void* const* d_in, const int* in_sizes, int n_in,
                              void* d_out, int out_size, void* d_ws, size_t ws_size,
                              hipStream_t stream) {
    const float* x = (const float*)d_in[0];
    const float* y = (const float*)d_in[1];
    float* out      = (float*)d_out;
    float* partials = (float*)d_ws;

    const long long n  = (long long)in_sizes[0];       // 32*1*1024*1024 = 2^25
    const long long n4 = n >> 2;

    int blocks = 4096;                                 // 2^25/4 elems / (4096*256) = 8 f4/thread... 16 f4-pairs total
    long long needed = (n4 + 255) / 256;
    if (needed < (long long)blocks) blocks = (int)(needed > 0 ? needed : 1);

    asym_l1_stage1<<<blocks, 256, 0, stream>>>(x, y, partials, n);
    asym_l1_stage2<<<1, 256, 0, stream>>>(partials, blocks, out, 1.0f / (float)n);
}